// Sparse_58660663329371
// MI455X (gfx1250) — compile-verified
//
#include <hip/hip_runtime.h>
#include <cstdint>

// Banded per-row affine map:
//   y[b,j] = x[b,j]*P[j] + x[b,j-1]*Q[j] + B[j]   (wrap x[-1] -> x[783])
// with extra terms x[b,782]*W[782,c] added to columns j=c (c=0,1),
// replicating the reference's wraparound/cutoff quirks exactly.
//
// Memory-bound (~1 FLOP/byte): stream x via CDNA5 async global->LDS DMA,
// compute from LDS (neighbor reuse + wrap reads are free), store y coalesced.

#define IN_N    784
#define ROWS    8            // batch rows per block
#define THREADS 256          // 8 wave32
#define CHUNKS  (IN_N / 4)   // 196 float4 chunks per row
#define TOTAL   (ROWS * CHUNKS)  // 1568

__global__ __launch_bounds__(THREADS) void sparse_banded_kernel(
    const float* __restrict__ x,   // [B, 784]
    const float* __restrict__ Wg,  // [784, 2] row-major: W[i*2+c]
    const float* __restrict__ bg,  // [784, 2]
    float* __restrict__ y)         // [B, 784]
{
    __shared__ float xs[ROWS * IN_N];   // staged x rows (25 KB)
    __shared__ float Pl[IN_N];          // coeff for x[j]
    __shared__ float Ql[IN_N];          // coeff for x[j-1] (wrap)
    __shared__ float Bl[IN_N];          // combined bias

    const int tid  = threadIdx.x;
    const int row0 = blockIdx.x * ROWS;

    // ---- Per-column coefficients (tiny, hits L2; recomputed per block) ----
    for (int j = tid; j < IN_N; j += THREADS) {
        float P, Q, Bv;
        if (j == 0) {                       // x0*W[0,0] + x783*W[783,1] (+x782 extra)
            P = Wg[0];       Q = Wg[1567];  Bv = bg[0] + bg[1564] + bg[1567];
        } else if (j == 1) {                // x1*W[1,0] + x0*W[0,1] (+x782 extra)
            P = Wg[2];       Q = Wg[1];     Bv = bg[2] + bg[1] + bg[1565];
        } else if (j == 782) {              // only x781*W[781,1]
            P = 0.0f;        Q = Wg[1563];  Bv = bg[1563];
        } else if (j == 783) {              // only x783*W[783,0]
            P = Wg[1566];    Q = 0.0f;      Bv = bg[1566];
        } else {                            // general band
            P = Wg[2 * j];   Q = Wg[2 * j - 1];
            Bv = bg[2 * j] + bg[2 * j - 1];
        }
        Pl[j] = P; Ql[j] = Q; Bl[j] = Bv;
    }

    // ---- Stage 8 rows of x into LDS via async DMA (ASYNCcnt-tracked) ----
    const float* xbase = x + (size_t)row0 * IN_N;
    #pragma unroll
    for (int k = 0; k < (TOTAL + THREADS - 1) / THREADS; ++k) {
        int idx = tid + k * THREADS;
        if (idx < TOTAL) {
            int row   = idx / CHUNKS;
            int chunk = idx - row * CHUNKS;
            const float* g = xbase + row * IN_N + chunk * 4;
            uint32_t lds_addr = (uint32_t)(uintptr_t)&xs[row * IN_N + chunk * 4];
            asm volatile("global_load_async_to_lds_b128 %0, %1, off"
                         :: "v"(lds_addr), "v"(g)
                         : "memory");
        }
    }
    asm volatile("s_wait_asynccnt 0" ::: "memory");
    __syncthreads();   // coefficients + all waves' async writes visible

    const float w782_0 = Wg[1564];  // W[782,0] -> extra into col 0
    const float w782_1 = Wg[1565];  // W[782,1] -> extra into col 1

    // ---- Compute + coalesced float4 stores ----
    #pragma unroll
    for (int k = 0; k < (TOTAL + THREADS - 1) / THREADS; ++k) {
        int idx = tid + k * THREADS;
        if (idx >= TOTAL) break;
        int row   = idx / CHUNKS;
        int chunk = idx - row * CHUNKS;
        int j0    = chunk * 4;

        const float* xr = &xs[row * IN_N];
        float4 A  = *reinterpret_cast<const float4*>(xr + j0);  // x[j0..j0+3]
        float xm1 = xr[(j0 == 0) ? 783 : (j0 - 1)];             // x[j0-1] (wrap)

        float4 P4 = *reinterpret_cast<const float4*>(&Pl[j0]);
        float4 Q4 = *reinterpret_cast<const float4*>(&Ql[j0]);
        float4 B4 = *reinterpret_cast<const float4*>(&Bl[j0]);

        float4 r;
        r.x = fmaf(A.x, P4.x, fmaf(xm1, Q4.x, B4.x));
        r.y = fmaf(A.y, P4.y, fmaf(A.x, Q4.y, B4.y));
        r.z = fmaf(A.z, P4.z, fmaf(A.y, Q4.z, B4.z));
        r.w = fmaf(A.w, P4.w, fmaf(A.z, Q4.w, B4.w));

        if (j0 == 0) {  // i=782 cutoff quirk: contributes to columns 0 and 1
            float x782 = xr[782];
            r.x = fmaf(x782, w782_0, r.x);
            r.y = fmaf(x782, w782_1, r.y);
        }

        *reinterpret_cast<float4*>(y + (size_t)(row0 + row) * IN_N + j0) = r;
    }
}

extern "C" void kernel_launch(void* const* d_in, const int* in_sizes, int n_in,
                              void* d_out, int out_size, void* d_ws, size_t ws_size,
                              hipStream_t stream) {
    (void)in_sizes; (void)n_in; (void)d_ws; (void)ws_size;
    const float* x  = (const float*)d_in[0];   // [32768, 784] f32
    const float* Wg = (const float*)d_in[1];   // [784, 2]    f32
    const float* bg = (const float*)d_in[2];   // [784, 2]    f32
    float* y = (float*)d_out;                  // [32768, 784] f32

    const int batch  = out_size / IN_N;        // 32768
    const int blocks = batch / ROWS;           // 4096
    sparse_banded_kernel<<<blocks, THREADS, 0, stream>>>(x, Wg, bg, y);
}